// FeatureDistributionConsistency_15745350107415
// MI455X (gfx1250) — compile-verified
//
#include <hip/hip_runtime.h>
#include <hip/hip_bf16.h>

typedef __attribute__((ext_vector_type(2))) float v2f;
typedef __attribute__((ext_vector_type(8))) float v8f;

#define N_SAMPLES   8192
#define NUM_CLASSES 64
#define FEAT_DIM    128
#define SLICE       288   // LDS dwords per k-pair slice (256 data + 32 pad):
                          // hi=0/hi=1 halves of a ds_load_b64 hit disjoint bank halves

// ---------------------------------------------------------------------------
// Stage 1: per-class Cholesky + triangular inverse + t = L^-1 mu.
// One block per class, 128 threads (row per thread), everything in LDS.
// Outputs (pair-interleaved B layout, ready for straight async copy to LDS):
//   linvT[c][(k>>1)*256 + 2n + (k&1)] = W[n_out=k? ...] -- element B[k][n] = W[n][k]
//   tvec[c][i] = (W mu_c)[i]
// ---------------------------------------------------------------------------
__global__ __launch_bounds__(128)
void chol_inv_kernel(const float* __restrict__ sigma,
                     const float* __restrict__ mu,
                     float* __restrict__ linvT,
                     float* __restrict__ tvec) {
  __shared__ float S[128 * 129];   // padded stride 129 -> conflict-free column reads
  __shared__ float col[128];
  __shared__ float wdiag[128];

  const int c   = blockIdx.x;
  const int tid = threadIdx.x;
  const float* sg = sigma + (size_t)c * FEAT_DIM * FEAT_DIM;

  for (int idx = tid; idx < 128 * 128; idx += 128) {
    int i = idx >> 7, j = idx & 127;
    S[i * 129 + j] = sg[idx];
  }
  __syncthreads();

  // Right-looking Cholesky, row per thread. After loop, lower(S) = L.
  for (int k = 0; k < 128; ++k) {
    float inv = 1.0f / sqrtf(S[k * 129 + k]);
    float cv  = (tid >= k) ? S[tid * 129 + k] * inv : 0.0f;
    col[tid] = cv;
    __syncthreads();
    for (int j = k + 1; j <= tid; ++j)
      S[tid * 129 + j] -= cv * col[j];
    S[tid * 129 + k] = cv;
    __syncthreads();
  }

  // Triangular inverse W = L^-1 (column per thread). W diag in wdiag[],
  // strict-lower W[i][j] (i>j) packed into strict-upper S at S[j*129+i].
  {
    const int cc = tid;
    float wd = 1.0f / S[cc * 129 + cc];
    wdiag[cc] = wd;
    for (int i = cc + 1; i < 128; ++i) {
      float s = S[i * 129 + cc] * wd;
      for (int j = cc + 1; j < i; ++j)
        s += S[i * 129 + j] * S[cc * 129 + j];
      S[cc * 129 + i] = -s / S[i * 129 + i];
    }
  }
  __syncthreads();

  // t[i] = sum_{j<=i} W[i][j] * mu[j]
  {
    const float* muc = mu + c * FEAT_DIM;
    float s = wdiag[tid] * muc[tid];
    for (int j = 0; j < tid; ++j)
      s += S[j * 129 + tid] * muc[j];
    tvec[c * FEAT_DIM + tid] = s;
  }

  // Emit B matrix, pair-interleaved: element B[k][n] = W[n][k] at
  // lt[(k>>1)*256 + 2n + (k&1)]  (tid == n, loop over k).
  float* lt = linvT + (size_t)c * FEAT_DIM * FEAT_DIM;
  for (int k = 0; k < 128; ++k) {
    float v = (tid > k) ? S[k * 129 + tid] : (tid == k ? wdiag[k] : 0.0f);
    lt[(k >> 1) * 256 + 2 * tid + (k & 1)] = v;
  }
}

// ---------------------------------------------------------------------------
// Async global->LDS staging of one class's B matrix (64 KB) using the CDNA5
// ASYNCcnt path. Straight chunk copy; LDS slices padded to SLICE dwords.
// Each of 256 threads moves 16 x 16B chunks.
// ---------------------------------------------------------------------------
__device__ __forceinline__ void stage_b_async(float* dst, const float* src, int tid) {
#pragma unroll
  for (int it = 0; it < 16; ++it) {
    const int q  = tid + it * 256;      // 16B-chunk index, 4096 total
    const int kp = q >> 6;              // k-pair slice
    const int ch = q & 63;              // chunk within slice
    unsigned ldsOff = (unsigned)(uintptr_t)(&dst[kp * SLICE + ch * 4]);
    unsigned long long ga = (unsigned long long)(uintptr_t)(src + q * 4);
    asm volatile("global_load_async_to_lds_b128 %0, %1, off"
                 :
                 : "v"(ldsOff), "v"(ga)
                 : "memory");
  }
}

__device__ __forceinline__ void wait_async_all() {
  asm volatile("s_wait_asynccnt 0x0" ::: "memory");
}

// ---------------------------------------------------------------------------
// Stage 2: Y = X * W^T via v_wmma_f32_16x16x4_f32 with double-buffered async
// B staging. Fused epilogue M[n,c] = sum_i (Y[n,i] - t_c[i])^2, reduced to
// per-workgroup (total, own-class) partials. Deterministic (no atomics).
// Grid: (N/128, 4 class groups), 256 threads = 8 waves, wave owns 16 rows.
// ---------------------------------------------------------------------------
__global__ __launch_bounds__(256)
void mahal_kernel(const float* __restrict__ X,
                  const int*   __restrict__ labels,
                  const float* __restrict__ linvT,
                  const float* __restrict__ tvec,
                  float* __restrict__ partials,
                  int classes_per_group) {
  __shared__ float Bs[2][64 * SLICE];   // 2 x 72 KB
  __shared__ float tsb[2][128];
  __shared__ int   ls[128];
  __shared__ float redT[16];
  __shared__ float redO[16];

  const int tid  = threadIdx.x;
  const int wave = tid >> 5;
  const int lane = tid & 31;
  const int l16  = lane & 15;
  const int hi   = lane >> 4;
  const int rowblk  = blockIdx.x;                 // 0..63
  const int cgrp    = blockIdx.y;                 // 0..3
  const int rowbase = rowblk * 128 + wave * 16;
  const int cpg     = classes_per_group;
  const int c0      = cgrp * cpg;

  if (tid < 128) ls[tid] = labels[rowblk * 128 + tid];

  // Preload A fragments: this wave's 16x128 X tile (ISA A layout:
  // lanes 0-15 hold M; VGPR pair = K {0,1} lanes 0-15 / {2,3} lanes 16-31).
  v2f a[32];
  {
    const float* xr = X + (size_t)(rowbase + l16) * FEAT_DIM + 2 * hi;
#pragma unroll
    for (int ks = 0; ks < 32; ++ks) {
      a[ks].x = xr[ks * 4 + 0];
      a[ks].y = xr[ks * 4 + 1];
    }
  }

  // Prologue: stage class c0 into buffer 0.
  stage_b_async(&Bs[0][0], linvT + (size_t)c0 * FEAT_DIM * FEAT_DIM, tid);
  if (tid < 128) tsb[0][tid] = tvec[c0 * FEAT_DIM + tid];
  wait_async_all();
  __syncthreads();

  float ctot = 0.0f, cown = 0.0f;

  for (int ci = 0; ci < cpg; ++ci) {
    const int c   = c0 + ci;
    const int cur = ci & 1;
    const float* Bcur = &Bs[cur][0];
    const float* tcur = &tsb[cur][0];

    // Kick off next class's async copy into the other buffer (its last
    // readers finished before the barrier that ended the previous iteration).
    if (ci + 1 < cpg) {
      stage_b_async(&Bs[cur ^ 1][0],
                    linvT + (size_t)(c + 1) * FEAT_DIM * FEAT_DIM, tid);
      if (tid < 128) tsb[cur ^ 1][tid] = tvec[(c + 1) * FEAT_DIM + tid];
    }

    float rowacc[8];
#pragma unroll
    for (int r = 0; r < 8; ++r) rowacc[r] = 0.0f;

#pragma unroll 1
    for (int nt = 0; nt < 8; ++nt) {
      const int nb = nt * 16 + l16;
      const float* bbase = Bcur + hi * SLICE + 2 * nb;

      // Batch all 32 B fragments (single ds_load_b64 each), then run the
      // 32 WMMAs back-to-back -> one DS wait per N-tile.
      v2f b[32];
#pragma unroll
      for (int ks = 0; ks < 32; ++ks)
        b[ks] = *(const v2f*)(bbase + ks * (2 * SLICE));

      v8f acc = {0.f, 0.f, 0.f, 0.f, 0.f, 0.f, 0.f, 0.f};
#pragma unroll
      for (int ks = 0; ks < 32; ++ks)
        acc = __builtin_amdgcn_wmma_f32_16x16x4_f32(
            false, a[ks], false, b[ks], (short)0, acc, false, false);

      // Epilogue: (Y - t)^2 accumulated per output row held by this lane.
      const float tc = tcur[nb];
#pragma unroll
      for (int r = 0; r < 8; ++r) {
        float v = acc[r] - tc;
        rowacc[r] += v * v;
      }
    }

    // Reduce across the 16-lane halves (D layout: row = r + 8*hi, col = l16).
#pragma unroll
    for (int r = 0; r < 8; ++r) {
      float v = rowacc[r];
      v += __shfl_xor(v, 1, 32);
      v += __shfl_xor(v, 2, 32);
      v += __shfl_xor(v, 4, 32);
      v += __shfl_xor(v, 8, 32);
      rowacc[r] = v;
    }
    if (l16 == 0) {
#pragma unroll
      for (int r = 0; r < 8; ++r) {
        const int rl = wave * 16 + 8 * hi + r;
        const float Mv = rowacc[r];
        ctot += Mv;
        if (ls[rl] == c) cown += Mv;
      }
    }

    // Drain this iteration's async copies, then barrier so (a) next-class
    // data is visible to all waves and (b) the buffer we will overwrite in
    // the next iteration is no longer being read.
    if (ci + 1 < cpg) wait_async_all();
    __syncthreads();
  }

  if (l16 == 0) { redT[wave * 2 + hi] = ctot; redO[wave * 2 + hi] = cown; }
  __syncthreads();
  if (tid == 0) {
    float T = 0.0f, O = 0.0f;
    for (int i = 0; i < 16; ++i) { T += redT[i]; O += redO[i]; }
    const int wg = blockIdx.y * gridDim.x + blockIdx.x;
    partials[2 * wg + 0] = T;
    partials[2 * wg + 1] = O;
  }
}

// ---------------------------------------------------------------------------
// Stage 3: deterministic fixed-order reduction of workgroup partials.
// ---------------------------------------------------------------------------
__global__ __launch_bounds__(256)
void finalize_kernel(const float* __restrict__ partials, int nwg,
                     float* __restrict__ out) {
  __shared__ float sT[256];
  __shared__ float sO[256];
  const int tid = threadIdx.x;
  float T = 0.0f, O = 0.0f;
  for (int i = tid; i < nwg; i += 256) { T += partials[2 * i]; O += partials[2 * i + 1]; }
  sT[tid] = T; sO[tid] = O;
  __syncthreads();
  for (int s = 128; s > 0; s >>= 1) {
    if (tid < s) { sT[tid] += sT[tid + s]; sO[tid] += sO[tid + s]; }
    __syncthreads();
  }
  if (tid == 0) {
    const float Tt = sT[0], Ot = sO[0];
    const float attraction = Ot / (float)N_SAMPLES;
    const float repulsion  = (Tt - Ot) / ((float)N_SAMPLES * (float)(NUM_CLASSES - 1));
    out[0] = repulsion - attraction;
  }
}

// ---------------------------------------------------------------------------
extern "C" void kernel_launch(void* const* d_in, const int* in_sizes, int n_in,
                              void* d_out, int out_size, void* d_ws, size_t ws_size,
                              hipStream_t stream) {
  (void)in_sizes; (void)n_in; (void)out_size; (void)ws_size;
  const float* features = (const float*)d_in[0];
  const float* mu       = (const float*)d_in[1];
  const float* sigma    = (const float*)d_in[2];
  const int*   labels   = (const int*)d_in[3];
  float* out = (float*)d_out;

  float* linvT    = (float*)d_ws;                                      // 4 MB
  float* tvec     = linvT + (size_t)NUM_CLASSES * FEAT_DIM * FEAT_DIM; // 32 KB
  float* partials = tvec + NUM_CLASSES * FEAT_DIM;                     // 2 KB

  chol_inv_kernel<<<NUM_CLASSES, 128, 0, stream>>>(sigma, mu, linvT, tvec);

  dim3 grid(N_SAMPLES / 128, 4);
  mahal_kernel<<<grid, 256, 0, stream>>>(features, labels, linvT, tvec,
                                         partials, NUM_CLASSES / 4);

  finalize_kernel<<<1, 256, 0, stream>>>(partials, (N_SAMPLES / 128) * 4, out);
}